// SparseConv_58162447123004
// MI455X (gfx1250) — compile-verified
//
#include <hip/hip_runtime.h>
#include <hip/hip_bf16.h>

typedef __attribute__((ext_vector_type(2))) float v2f;
typedef __attribute__((ext_vector_type(8))) float v8f;

#define HEIGHT 480
#define WIDTH  640
#define H1 481
#define W1 641
#define CIN 8
#define COUT 32
#define NB 8
#define NPTS 16384
#define TOTAL_PTS (NB * NPTS)
#define GRID_FLOATS (NB * H1 * W1 * CIN)   // 19,732,544 floats = ~75.3 MiB

// ---------------------------------------------------------------- zero grid
__global__ void SparseConv_zero_kernel(float4* __restrict__ ws, int n4) {
    int i = blockIdx.x * blockDim.x + threadIdx.x;
    int stride = gridDim.x * blockDim.x;
    float4 z = {0.f, 0.f, 0.f, 0.f};
    for (; i < n4; i += stride) ws[i] = z;
}

// ---------------------------------------------------------------- scatter-add
__global__ void SparseConv_scatter_kernel(const float* __restrict__ xytp,
                                          const float* __restrict__ feats,
                                          float* __restrict__ grid) {
    int g = blockIdx.x * blockDim.x + threadIdx.x;
    if (g >= TOTAL_PTS) return;
    int b = g >> 14;                       // N = 16384
    const float* xp = xytp + (size_t)g * 4;
    float x = xp[1], y = xp[2], p = xp[3];
    int xx = __float2int_rn(x * (float)WIDTH);    // round-to-nearest-even == jnp.round
    int yy = __float2int_rn(y * (float)HEIGHT);
    size_t base = ((size_t)(b * H1 + yy) * W1 + xx) * CIN;
    const float* f = feats + (size_t)g * 6;
    float v[8] = {p, 1.0f - p, f[0], f[1], f[2], f[3], f[4], f[5]};
#pragma unroll
    for (int c = 0; c < 8; ++c)
        unsafeAtomicAdd(&grid[base + c], v[c]);   // global_atomic_add_f32
}

// ---------------------------------------------------------------- per-point conv via f32 WMMA
// One wave (32 lanes) computes 16 points x 32 Cout:
//   D(16x32) = patch(16x72) * W(72x32) + bias, as 2 N-tiles, 18 K=4 WMMA steps.
__global__ __launch_bounds__(32)
void SparseConv_wmma_kernel(const float* __restrict__ xytp,
                            const float* __restrict__ grid,
                            const float* __restrict__ Wt,      // (3,3,8,32) row-major -> [j][co], j=(ky*3+kx)*8+ci
                            const float* __restrict__ bias,
                            float* __restrict__ out) {
    __shared__ float patch[16][76];   // stride 76 floats: conflict-free ds_load_b64
    __shared__ int px[16], py[16], pb[16];

    const int lane = threadIdx.x;
    const int p0 = blockIdx.x * 16;

    if (lane < 16) {
        int g = p0 + lane;
        const float* xp = xytp + (size_t)g * 4;
        px[lane] = __float2int_rn(xp[1] * (float)WIDTH);
        py[lane] = __float2int_rn(xp[2] * (float)HEIGHT);
        pb[lane] = g >> 14;
    }
    __syncthreads();

    // 16 points x 9 taps = 144 tasks; each task pulls 8 contiguous channel floats.
    for (int i = lane; i < 144; i += 32) {
        int pt  = i / 9;
        int tap = i - pt * 9;
        int y = py[pt] + tap / 3 - 1;
        int x = px[pt] + tap % 3 - 1;
        float4 v0 = {0.f, 0.f, 0.f, 0.f}, v1 = v0;   // SAME padding -> zeros OOB
        if ((unsigned)y < (unsigned)H1 && (unsigned)x < (unsigned)W1) {
            const float4* src =
                (const float4*)(grid + ((size_t)(pb[pt] * H1 + y) * W1 + x) * CIN);
            v0 = src[0];
            v1 = src[1];
        }
        float4* dst = (float4*)(&patch[pt][tap * 8]);
        dst[0] = v0;
        dst[1] = v1;
    }
    __syncthreads();

    // A-frag (16x4 f32): lanes 0-15 row m, K={j0,j0+1}; lanes 16-31 row m, K={j0+2,j0+3}
    // B-frag (4x16 f32): mirrored K striping, N = lane&15
    const int   m    = lane & 15;
    const int   koff = (lane < 16) ? 0 : 2;
    const float* prow = &patch[m][0];

    v8f c0 = {0.f, 0.f, 0.f, 0.f, 0.f, 0.f, 0.f, 0.f};
    v8f c1 = c0;

#pragma unroll
    for (int k = 0; k < 18; ++k) {
        int j0 = 4 * k + koff;
        v2f a;
        a.x = prow[j0];
        a.y = prow[j0 + 1];
        const float* w0 = Wt + j0 * COUT + m;
        v2f b0, b1;
        b0.x = w0[0];        b0.y = w0[COUT];        // Cout tile 0 (cols 0-15)
        b1.x = w0[16];       b1.y = w0[COUT + 16];   // Cout tile 1 (cols 16-31)
        c0 = __builtin_amdgcn_wmma_f32_16x16x4_f32(false, a, false, b0, (short)0, c0, false, false);
        c1 = __builtin_amdgcn_wmma_f32_16x16x4_f32(false, a, false, b1, (short)0, c1, false, false);
    }

    // C/D layout: VGPR r -> M = r (lanes 0-15) or r+8 (lanes 16-31); N = lane&15.
    const int rowoff = (lane < 16) ? 0 : 8;
    float bl = bias[m], bh = bias[m + 16];
#pragma unroll
    for (int r = 0; r < 8; ++r) {
        size_t row = (size_t)(p0 + r + rowoff);
        out[row * COUT + m]      = c0[r] + bl;
        out[row * COUT + 16 + m] = c1[r] + bh;
    }
}

// ---------------------------------------------------------------- launch
extern "C" void kernel_launch(void* const* d_in, const int* in_sizes, int n_in,
                              void* d_out, int out_size, void* d_ws, size_t ws_size,
                              hipStream_t stream) {
    (void)in_sizes; (void)n_in; (void)out_size; (void)ws_size;
    const float* xytp  = (const float*)d_in[0];
    const float* feats = (const float*)d_in[1];
    const float* Wt    = (const float*)d_in[2];
    const float* bias  = (const float*)d_in[3];
    float* grid = (float*)d_ws;              // needs GRID_FLOATS*4 = ~75.3 MiB of scratch
    float* out  = (float*)d_out;

    SparseConv_zero_kernel<<<2048, 256, 0, stream>>>((float4*)grid, GRID_FLOATS / 4);
    SparseConv_scatter_kernel<<<TOTAL_PTS / 256, 256, 0, stream>>>(xytp, feats, grid);
    SparseConv_wmma_kernel<<<TOTAL_PTS / 16, 32, 0, stream>>>(xytp, grid, Wt, bias, out);
}